// GATEAULayer_5282809774964
// MI455X (gfx1250) — compile-verified
//
#include <hip/hip_runtime.h>
#include <math.h>
#include <stdint.h>

// ---------------- problem constants (from reference) ----------------
#define N_NODES 50000
#define N_EDGES 600000
#define D 128            // all feature dims are 128; divisible by 16 tiles
#define LDS_PITCH 132    // 132*4B = 528B row pitch: 16B-aligned (b128 async) and
                         // bank-stride 4 -> conflict-free A-operand reads

typedef __attribute__((ext_vector_type(2))) float v2f;
typedef __attribute__((ext_vector_type(8))) float v8f;

// =====================================================================
// Async DMA of a 16x128 fp32 tile into LDS (CDNA5 global_load_async_to_lds,
// ASYNCcnt-tracked, no VGPR round trip). 256 threads x 2 x b128 = 8KB.
// =====================================================================
__device__ __forceinline__ void tile_load_async(float (*Xs)[LDS_PITCH],
                                                const float* __restrict__ gsrc,
                                                int row0, int tid) {
    #pragma unroll
    for (int i = 0; i < 2; ++i) {
        int idx4 = tid + i * 256;            // 512 float4 chunks: 16 rows x 32
        int r  = idx4 >> 5;
        int c4 = idx4 & 31;
        unsigned lds_addr = (unsigned)(uintptr_t)(&Xs[r][c4 * 4]);  // low 32b = LDS offset
        unsigned long long gaddr =
            (unsigned long long)(uintptr_t)(gsrc + (size_t)(row0 + r) * D + c4 * 4);
        asm volatile("global_load_async_to_lds_b128 %0, %1, off"
                     :: "v"(lds_addr), "v"(gaddr) : "memory");
    }
    asm volatile("s_wait_asynccnt 0" ::: "memory");
}

// =====================================================================
// One 16x16 fp32 output tile: K=128 swept as 32 chained
// V_WMMA_F32_16X16X4_F32. A from LDS tile, B (weights) from global (L0/L2
// resident). ISA 16x4 f32 A layout: lanes 0-15 = M, v0/v1 = K,K+1; lanes
// 16-31 = K+2,K+3. C/D: VGPR r -> rows r / r+8.
// =====================================================================
__device__ __forceinline__ v8f wmma_tile(const float Xs[][LDS_PITCH],
                                         const float* __restrict__ W,
                                         int m, int kh, int n0) {
    v8f acc = {0.f, 0.f, 0.f, 0.f, 0.f, 0.f, 0.f, 0.f};
    #pragma unroll
    for (int k0 = 0; k0 < D; k0 += 4) {
        v2f a, b;
        a.x = Xs[m][k0 + 2 * kh];
        a.y = Xs[m][k0 + 2 * kh + 1];
        b.x = W[(size_t)(k0 + 2 * kh)     * D + n0 + m];
        b.y = W[(size_t)(k0 + 2 * kh + 1) * D + n0 + m];
        acc = __builtin_amdgcn_wmma_f32_16x16x4_f32(
            /*neg_a=*/false, a, /*neg_b=*/false, b,
            /*c_mod=*/(short)0, acc, /*reuse_a=*/false, /*reuse_b=*/false);
    }
    return acc;
}

// =====================================================================
// Kernel 1: init segment-softmax state
// =====================================================================
__global__ void gateau_init_kernel(float* __restrict__ smax, float* __restrict__ ssum) {
    int i = blockIdx.x * blockDim.x + threadIdx.x;
    if (i < N_NODES) {
        ((int*)smax)[i] = 0xFF800000;   // -inf bits (int-max / uint-min trick)
        ssum[i] = 0.0f;
    }
}

// =====================================================================
// Kernel 2: fused node projections. One X tile in LDS feeds all FOUR
// weight matrices (X read once instead of 4x). 128 WMMAs per wave.
// h_0 lands directly in d_out (seed for the scatter pass).
// =====================================================================
__global__ void gateau_node_gemm_kernel(const float* __restrict__ X,
                                        const float* __restrict__ Wv,
                                        const float* __restrict__ Wu,
                                        const float* __restrict__ Wh,
                                        const float* __restrict__ W0,
                                        float* __restrict__ hv,
                                        float* __restrict__ hu,
                                        float* __restrict__ hh,
                                        float* __restrict__ out_nodes) {
    __shared__ float Xs[16][LDS_PITCH];
    const int tid  = threadIdx.x;
    const int row0 = blockIdx.x * 16;

    tile_load_async(Xs, X, row0, tid);
    __syncthreads();

    const int lane = tid & 31, wave = tid >> 5;
    const int m = lane & 15, kh = lane >> 4, n0 = wave * 16;

    const float* Ws[4] = {Wv, Wu, Wh, W0};
    float*       Ys[4] = {hv, hu, hh, out_nodes};
    #pragma unroll
    for (int mt = 0; mt < 4; ++mt) {
        v8f acc = wmma_tile(Xs, Ws[mt], m, kh, n0);
        #pragma unroll
        for (int r = 0; r < 8; ++r)
            Ys[mt][(size_t)(row0 + r + 8 * kh) * D + n0 + m] = acc[r];
    }
}

// =====================================================================
// Kernel 3 (edge pass 1): h_e = Ef@We, fused gather epilogue
// g_prime = h_e + h_u[tgt] + h_v[src] -> d_out, AND fused attention
// score: per-wave 16-lane shfl reduction of g_prime . a into an LDS
// accumulator (ds_add_f32), leaky-relu, write scores. Eliminates the
// 307MB g_prime re-read of the standalone score kernel.
// =====================================================================
__global__ void gateau_edge_pass1_kernel(const float* __restrict__ Ef,
                                         const float* __restrict__ We,
                                         const float* __restrict__ hu,
                                         const float* __restrict__ hv,
                                         const int*   __restrict__ tgt,
                                         const int*   __restrict__ src,
                                         const float* __restrict__ avec,
                                         float* __restrict__ gprime,
                                         float* __restrict__ scores) {
    __shared__ float Xs[16][LDS_PITCH];
    __shared__ float sacc[16];
    const int tid  = threadIdx.x;
    const int row0 = blockIdx.x * 16;

    if (tid < 16) sacc[tid] = 0.0f;
    tile_load_async(Xs, Ef, row0, tid);
    __syncthreads();

    const int lane = tid & 31, wave = tid >> 5;
    const int m = lane & 15, kh = lane >> 4, n0 = wave * 16;

    v8f acc = wmma_tile(Xs, We, m, kh, n0);

    const float ael = avec[n0 + m];
    #pragma unroll
    for (int r = 0; r < 8; ++r) {
        int e = row0 + r + 8 * kh;
        int t = tgt[e], s = src[e];
        float v = acc[r]
                + hu[(size_t)t * D + n0 + m]
                + hv[(size_t)s * D + n0 + m];
        gprime[(size_t)e * D + n0 + m] = v;

        // partial dot with 'a' over this wave's 16 columns; xor offsets
        // 8..1 stay inside each 16-lane half (wave32)
        float part = v * ael;
        #pragma unroll
        for (int off = 8; off > 0; off >>= 1)
            part += __shfl_xor(part, off, 32);
        if (m == 0) atomicAdd(&sacc[r + 8 * kh], part);
    }
    __syncthreads();

    if (tid < 16) {
        float s = sacc[tid];
        scores[row0 + tid] = s > 0.0f ? s : 0.2f * s;   // leaky_relu slope 0.2
    }
}

// =====================================================================
// Kernel 4: segment max over tgt (sign-split monotone-bits atomic trick)
// =====================================================================
__global__ void gateau_segmax_kernel(const float* __restrict__ scores,
                                     const int*   __restrict__ tgt,
                                     float* __restrict__ smax) {
    int e = blockIdx.x * blockDim.x + threadIdx.x;
    if (e >= N_EDGES) return;
    float s = scores[e];
    int   t = tgt[e];
    if (s >= 0.0f)
        atomicMax((int*)(smax + t), __float_as_int(s));
    else
        atomicMin((unsigned int*)(smax + t), (unsigned int)__float_as_int(s));
}

// =====================================================================
// Kernel 5: exp(score - max[tgt]) and segment sum
// =====================================================================
__global__ void gateau_expsum_kernel(const float* __restrict__ scores,
                                     const int*   __restrict__ tgt,
                                     const float* __restrict__ smax,
                                     float* __restrict__ exps,
                                     float* __restrict__ ssum) {
    int e = blockIdx.x * blockDim.x + threadIdx.x;
    if (e >= N_EDGES) return;
    int t = tgt[e];
    float v = expf(scores[e] - smax[t]);
    exps[e] = v;
    atomicAdd(ssum + t, v);
}

// =====================================================================
// Kernel 6 (edge pass 2): RECOMPUTE h_g = Ef@Wg via WMMA (re-reading Ef
// once is 307MB cheaper than writing+reading an h_g buffer), then the
// alpha-weighted scatter: out[tgt] += (h_h[src] + h_g) * alpha.
// out_nodes was seeded with h_0 by the node GEMM.
// =====================================================================
__global__ void gateau_edge_pass2_kernel(const float* __restrict__ Ef,
                                         const float* __restrict__ Wg,
                                         const float* __restrict__ hh,
                                         const int*   __restrict__ tgt,
                                         const int*   __restrict__ src,
                                         const float* __restrict__ exps,
                                         const float* __restrict__ ssum,
                                         float* __restrict__ out_nodes) {
    __shared__ float Xs[16][LDS_PITCH];
    const int tid  = threadIdx.x;
    const int row0 = blockIdx.x * 16;

    tile_load_async(Xs, Ef, row0, tid);
    __syncthreads();

    const int lane = tid & 31, wave = tid >> 5;
    const int m = lane & 15, kh = lane >> 4, n0 = wave * 16;

    v8f acc = wmma_tile(Xs, Wg, m, kh, n0);

    #pragma unroll
    for (int r = 0; r < 8; ++r) {
        int e = row0 + r + 8 * kh;
        int t = tgt[e], s = src[e];
        float alpha = exps[e] / (ssum[t] + 1e-10f);
        float w = (hh[(size_t)s * D + n0 + m] + acc[r]) * alpha;
        atomicAdd(&out_nodes[(size_t)t * D + n0 + m], w);
    }
}

// =====================================================================
// host-side launcher
// =====================================================================
extern "C" void kernel_launch(void* const* d_in, const int* in_sizes, int n_in,
                              void* d_out, int out_size, void* d_ws, size_t ws_size,
                              hipStream_t stream) {
    const float* X  = (const float*)d_in[0];   // node_feature_matrix [N,128]
    const float* Ef = (const float*)d_in[1];   // edge_feature_matrix [E,128]
    const int*   ei = (const int*)  d_in[2];   // edge_index [2,E]
    const float* Wv = (const float*)d_in[3];
    const float* Wu = (const float*)d_in[4];
    const float* We = (const float*)d_in[5];
    const float* Wh = (const float*)d_in[6];
    const float* Wg = (const float*)d_in[7];
    const float* W0 = (const float*)d_in[8];
    const float* a  = (const float*)d_in[9];
    (void)in_sizes; (void)n_in; (void)out_size; (void)ws_size;

    const int* tgt = ei;
    const int* src = ei + N_EDGES;

    // output: [node_out (N*128) | g_prime (E*128)]
    float* out_nodes = (float*)d_out;
    float* gprime    = out_nodes + (size_t)N_NODES * D;

    // workspace layout (floats): hv, hu, hh, scores, exps, smax, ssum (~82MB)
    float* ws     = (float*)d_ws;
    float* hv     = ws;
    float* hu     = hv + (size_t)N_NODES * D;
    float* hh     = hu + (size_t)N_NODES * D;
    float* scores = hh + (size_t)N_NODES * D;
    float* exps   = scores + N_EDGES;
    float* smax   = exps + N_EDGES;
    float* ssum   = smax + N_NODES;

    gateau_init_kernel<<<(N_NODES + 255) / 256, 256, 0, stream>>>(smax, ssum);

    gateau_node_gemm_kernel<<<N_NODES / 16, 256, 0, stream>>>(
        X, Wv, Wu, Wh, W0, hv, hu, hh, out_nodes);

    gateau_edge_pass1_kernel<<<N_EDGES / 16, 256, 0, stream>>>(
        Ef, We, hu, hv, tgt, src, a, gprime, scores);

    gateau_segmax_kernel<<<(N_EDGES + 255) / 256, 256, 0, stream>>>(
        scores, tgt, smax);

    gateau_expsum_kernel<<<(N_EDGES + 255) / 256, 256, 0, stream>>>(
        scores, tgt, smax, exps, ssum);

    gateau_edge_pass2_kernel<<<N_EDGES / 16, 256, 0, stream>>>(
        Ef, Wg, hh, tgt, src, exps, ssum, out_nodes);
}